// SCentroidDistance_2954937500465
// MI455X (gfx1250) — compile-verified
//
#include <hip/hip_runtime.h>
#include <math.h>

typedef float v2f __attribute__((ext_vector_type(2)));
typedef float v8f __attribute__((ext_vector_type(8)));

#define N_NODES 8192
#define K_CENT  256
#define D_DIM   128
#define EPSF    1e-15f
#define PROJ_EPS 4e-3f

#define BM 64          // node rows per block
#define BN 32          // centroid cols per block
#define PITCH 132      // D_DIM + 4 pad floats: row stride % 64 banks == 4 -> conflict-free

// workspace layout (float offsets)
#define OFF_YPROJ   0                         // 256*128
#define OFF_Y2      (K_CENT * D_DIM)          // 256
#define OFF_X2      (OFF_Y2 + K_CENT)         // 8192
#define OFF_COLACC  (OFF_X2 + N_NODES)        // 256
#define OFF_MASKSUM (OFF_COLACC + K_CENT)     // 1

// ---- CDNA5 async global->LDS staging (ASYNCcnt-tracked), with safe fallback ----
#if defined(__has_builtin)
#if __has_builtin(__builtin_amdgcn_global_load_async_to_lds_b128)
#define HAVE_ASYNC_LDS 1
#endif
#endif

// exact parameter pointee type per hipcc diagnostic: GCC-vector int4
typedef int v4i_b __attribute__((vector_size(4 * sizeof(int))));
typedef __attribute__((address_space(1))) v4i_b* gas_v4i_p;
typedef __attribute__((address_space(3))) v4i_b* las_v4i_p;

__device__ __forceinline__ void stage16(const float* __restrict__ g, float* l) {
#ifdef HAVE_ASYNC_LDS
    __builtin_amdgcn_global_load_async_to_lds_b128((gas_v4i_p)g, (las_v4i_p)l, 0, 0);
#else
    *(float4*)l = *(const float4*)g;
#endif
}

__device__ __forceinline__ void stage_wait() {
#ifdef HAVE_ASYNC_LDS
#if defined(__has_builtin) && __has_builtin(__builtin_amdgcn_s_wait_asynccnt)
    __builtin_amdgcn_s_wait_asynccnt(0);
#else
    asm volatile("s_wait_asynccnt 0x0" ::: "memory");
#endif
#endif
}

// ---------------- prep: centroid projection, y2, mask sum, zero accumulators ----------------
__global__ __launch_bounds__(256) void prep_centroids(const float* __restrict__ cw,
                                                      const float* __restrict__ mask,
                                                      const float* __restrict__ cvec,
                                                      float* __restrict__ ws) {
    float* yproj   = ws + OFF_YPROJ;
    float* y2      = ws + OFF_Y2;
    float* colacc  = ws + OFF_COLACC;
    float* masksum = ws + OFF_MASKSUM;

    const int t = threadIdx.x;               // 256 threads, 1 block; thread t owns centroid t
    const float cc = cvec[0];
    const float maxnorm = (1.0f - PROJ_EPS) / sqrtf(cc);

    const float* row = cw + t * D_DIM;
    float n2 = 0.0f;
    for (int d = 0; d < D_DIM; d += 4) {
        float4 v = *(const float4*)(row + d);
        n2 += v.x * v.x + v.y * v.y + v.z * v.z + v.w * v.w;
    }
    float nrm = fmaxf(sqrtf(n2), EPSF);
    float scale = (nrm > maxnorm) ? (maxnorm / nrm) : 1.0f;
    for (int d = 0; d < D_DIM; d += 4) {
        float4 v = *(const float4*)(row + d);
        v.x *= scale; v.y *= scale; v.z *= scale; v.w *= scale;
        *(float4*)(yproj + t * D_DIM + d) = v;
    }
    float pn = fminf(nrm, maxnorm);
    y2[t] = pn * pn;
    colacc[t] = 0.0f;

    // mask sum over N (mask is [N,1])
    __shared__ float red[256];
    float s = 0.0f;
    for (int i = t; i < N_NODES; i += 256) s += mask[i];
    red[t] = s;
    __syncthreads();
    for (int off = 128; off > 0; off >>= 1) {
        if (t < off) red[t] += red[t + off];
        __syncthreads();
    }
    if (t == 0) masksum[0] = red[0];
}

// ---------------- prep: x2 per node ----------------
__global__ __launch_bounds__(256) void prep_x2(const float* __restrict__ X, float* __restrict__ ws) {
    float* x2 = ws + OFF_X2;
    const int n = blockIdx.x * 256 + threadIdx.x;
    const float* row = X + (size_t)n * D_DIM;
    float s = 0.0f;
    for (int d = 0; d < D_DIM; d += 4) {
        float4 v = *(const float4*)(row + d);
        s += v.x * v.x + v.y * v.y + v.z * v.z + v.w * v.w;
    }
    x2[n] = s;
}

// ---------------- main fused kernel: f32 WMMA GEMM + hyperbolic epilogue ----------------
__global__ __launch_bounds__(256) void poincare_dist_kernel(const float* __restrict__ X,
                                                            const float* __restrict__ cvec,
                                                            float* __restrict__ ws,
                                                            float* __restrict__ out) {
    const float* yproj = ws + OFF_YPROJ;
    const float* y2g   = ws + OFF_Y2;
    const float* x2g   = ws + OFF_X2;
    float* colacc      = ws + OFF_COLACC;

    __shared__ float sX[BM * PITCH];
    __shared__ float sY[BN * PITCH];
    __shared__ float sx2[BM];
    __shared__ float sy2[BN];
    __shared__ float scol[BN];

    const int tid   = threadIdx.x;
    const int mbase = blockIdx.x * BM;
    const int nbase = blockIdx.y * BN;

    // cooperative stage: X tile [BM x 128] and projected-Y tile [BN x 128] into LDS
    // via gfx1250 async global->LDS (16B per lane per op), coalesced along D
    for (int i = tid; i < BM * (D_DIM / 4); i += 256) {
        int r = i >> 5, cq = (i & 31) << 2;
        stage16(X + (size_t)(mbase + r) * D_DIM + cq, &sX[r * PITCH + cq]);
    }
    for (int i = tid; i < BN * (D_DIM / 4); i += 256) {
        int r = i >> 5, cq = (i & 31) << 2;
        stage16(yproj + (size_t)(nbase + r) * D_DIM + cq, &sY[r * PITCH + cq]);
    }
    if (tid < BM) sx2[tid] = x2g[mbase + tid];
    if (tid < BN) { sy2[tid] = y2g[nbase + tid]; scol[tid] = 0.0f; }
    stage_wait();            // drain this wave's ASYNCcnt before publishing LDS
    __syncthreads();

    const int wave = tid >> 5;     // 8 waves
    const int lane = tid & 31;
    const int half = lane >> 4;    // K-half selector per ISA 16x4 f32 layout
    const int l16  = lane & 15;
    const int msub = wave & 3;     // 4 M-subtiles of 16
    const int nsub = wave >> 2;    // 2 N-subtiles of 16

    const float* Arow = &sX[(msub * 16 + l16) * PITCH + 2 * half];
    const float* Brow = &sY[(nsub * 16 + l16) * PITCH + 2 * half];

    v8f acc = {};
#pragma unroll
    for (int k0 = 0; k0 < D_DIM; k0 += 4) {
        v2f a = *(const v2f*)(Arow + k0);   // lanes 0-15: K=k0,k0+1 ; lanes 16-31: K=k0+2,k0+3
        v2f b = *(const v2f*)(Brow + k0);
        acc = __builtin_amdgcn_wmma_f32_16x16x4_f32(false, a, false, b, (short)0, acc, false, false);
    }

    // epilogue: acc[r] = dot(x_node, y_cent) at (M = msub*16 + r + 8*half, N = nsub*16 + l16)
    const float cc = cvec[0];
    const float sqrt_c = sqrtf(cc);
    const float inv2 = 2.0f / sqrt_c;
    const float myY2 = sy2[nsub * 16 + l16];
    const int kcol = nbase + nsub * 16 + l16;
    float* outnode = out + K_CENT;   // node_centroid_dist region
    float colsum = 0.0f;
#pragma unroll
    for (int r = 0; r < 8; r++) {
        int mloc = msub * 16 + r + 8 * half;
        float x2v = sx2[mloc];
        float dot = acc[r];
        float xy = -dot;                                   // <-x, y>
        float Af = 1.0f + 2.0f * cc * xy + cc * myY2;
        float Bf = 1.0f - cc * x2v;
        float num2 = Af * Af * x2v + 2.0f * Af * Bf * xy + Bf * Bf * myY2;
        num2 = fmaxf(num2, 0.0f);
        float denom = 1.0f + 2.0f * cc * xy + cc * cc * x2v * myY2;
        denom = fmaxf(denom, EPSF);
        float dn = sqrtf(num2) / denom;                    // ||mobius_add(-x,y,c)||
        float aa = fminf(sqrt_c * dn, 1.0f - EPSF);
        float ath = 0.5f * (log1pf(aa) - log1pf(-aa));     // artanh
        float dist = inv2 * ath;
        float d2 = dist * dist;
        outnode[(size_t)(mbase + mloc) * K_CENT + kcol] = d2;
        colsum += d2;
    }
    // fold the two M-halves (lane l and l+16 share a column), then block-level column accumulate
    colsum += __shfl_xor(colsum, 16, 32);
    if (half == 0) atomicAdd(&scol[nsub * 16 + l16], colsum);
    __syncthreads();
    if (tid < BN) atomicAdd(&colacc[nbase + tid], scol[tid]);
}

// ---------------- finalize: graph_centroid_dist = colsum / sum(mask) ----------------
__global__ __launch_bounds__(256) void finalize_graph(const float* __restrict__ ws,
                                                      float* __restrict__ out) {
    const float* colacc  = ws + OFF_COLACC;
    const float* masksum = ws + OFF_MASKSUM;
    int k = threadIdx.x;
    out[k] = colacc[k] / masksum[0];
}

extern "C" void kernel_launch(void* const* d_in, const int* in_sizes, int n_in,
                              void* d_out, int out_size, void* d_ws, size_t ws_size,
                              hipStream_t stream) {
    (void)in_sizes; (void)n_in; (void)out_size; (void)ws_size;
    const float* node = (const float*)d_in[0];   // [8192,128]
    const float* mask = (const float*)d_in[1];   // [8192,1]
    const float* cw   = (const float*)d_in[2];   // [256,128]
    const float* c    = (const float*)d_in[3];   // [1]
    float* ws  = (float*)d_ws;
    float* out = (float*)d_out;                  // [256] graph dist, then [8192*256] node dist

    prep_centroids<<<1, 256, 0, stream>>>(cw, mask, c, ws);
    prep_x2<<<N_NODES / 256, 256, 0, stream>>>(node, ws);
    dim3 grid(N_NODES / BM, K_CENT / BN);        // 128 x 8 blocks
    poincare_dist_kernel<<<grid, 256, 0, stream>>>(node, c, ws, out);
    finalize_graph<<<1, K_CENT, 0, stream>>>(ws, out);
}